// EdgeGateProj_21947282882710
// MI455X (gfx1250) — compile-verified
//
#include <hip/hip_runtime.h>

typedef __attribute__((ext_vector_type(2))) float v2f;
typedef __attribute__((ext_vector_type(8))) float v8f;

#define FEAT_D 128

// Phase 1: per-node scores via FP32 WMMA.
// One wave32 handles 16 nodes. D = A(16x4) x B(4x16) + C, K swept 0..127 in
// steps of 4 (32 wmma ops, 2 interleaved accumulator chains for ILP).
//   A rows: M=0 -> w_src chunk, M=1 -> w_tgt chunk, M>=2 -> w_tgt junk
//           (harmless: D row m depends only on A row m; rows 2..15 discarded)
//   B cols: node features (transposed chunk): B[k][n] = x[node_n][kbase+k]
//   D[0][n] = s_src[node_n], D[1][n] = s_tgt[node_n]
__global__ __launch_bounds__(256) void node_scores_wmma_kernel(
    const float* __restrict__ x,
    const float* __restrict__ w_src,
    const float* __restrict__ w_tgt,
    float* __restrict__ s_src,
    float* __restrict__ s_tgt,
    int n_nodes, int n_tiles)
{
    const int wave = (int)((blockIdx.x * blockDim.x + threadIdx.x) >> 5);
    const int lane = (int)(threadIdx.x & 31);
    if (wave >= n_tiles) return;            // wave-uniform exit

    const int n   = lane & 15;              // A: row M / B: col N / D: col N
    const int hi  = lane >> 4;              // K-half selector (K = j + 2*hi)
    const int node  = wave * 16 + n;
    const int nodeC = (node < n_nodes) ? node : (n_nodes - 1);

    // B loads: contiguous float2 per lane -> global_load_b64
    const float* xrow = x + (size_t)nodeC * FEAT_D + 2 * hi;
    // A loads: unconditional for ALL lanes (no exec-mask churn).
    // Lanes n>=2 load w_tgt values; they only affect discarded D rows.
    const float* wrow = ((n == 0) ? w_src : w_tgt) + 2 * hi;

    v8f acc0 = {};
    v8f acc1 = {};
    #pragma unroll
    for (int c = 0; c < FEAT_D / 4; ++c) {
        const int k = 4 * c;
        v2f b = *(const v2f*)(xrow + k);    // B[k..k+1][n], K = 4c+2hi+{0,1}
        v2f a = *(const v2f*)(wrow + k);    // A[n][k..k+1], same K slots
        if (c & 1)
            acc1 = __builtin_amdgcn_wmma_f32_16x16x4_f32(
                false, a, false, b, (short)0, acc1, false, false);
        else
            acc0 = __builtin_amdgcn_wmma_f32_16x16x4_f32(
                false, a, false, b, (short)0, acc0, false, false);
    }

    // D VGPR p, lanes 0-15: M=p, N=lane  ->  acc[0]=s_src row, acc[1]=s_tgt row
    if (hi == 0 && node < n_nodes) {
        s_src[node] = acc0[0] + acc1[0];
        s_tgt[node] = acc0[1] + acc1[1];
    }
}

// Phase 2: per-edge gather of two L2-resident scalars + sigmoid.
__global__ __launch_bounds__(256) void edge_gate_kernel(
    const long long* __restrict__ edge_index,   // [2, E] int64
    const float* __restrict__ s_src,
    const float* __restrict__ s_tgt,
    float* __restrict__ out,
    int n_edges)
{
    const int e = (int)(blockIdx.x * blockDim.x + threadIdx.x);
    if (e >= n_edges) return;
    const long long si = edge_index[e];                       // src row
    const long long ti = edge_index[(size_t)n_edges + e];     // tgt row
    const float g = s_src[si] + s_tgt[ti];
    out[e] = 1.0f / (1.0f + expf(-g));
}

extern "C" void kernel_launch(void* const* d_in, const int* in_sizes, int n_in,
                              void* d_out, int out_size, void* d_ws, size_t ws_size,
                              hipStream_t stream) {
    const float*     x          = (const float*)d_in[0];
    const long long* edge_index = (const long long*)d_in[1];
    const float*     w_src      = (const float*)d_in[2];
    const float*     w_tgt      = (const float*)d_in[3];
    float*           out        = (float*)d_out;

    const int n_nodes = in_sizes[0] / FEAT_D;   // 100000
    const int n_edges = in_sizes[1] / 2;        // 1000000

    // scratch: per-node score arrays (2 * n_nodes * 4B = 800 KB)
    float* s_src = (float*)d_ws;
    float* s_tgt = s_src + n_nodes;

    const int n_tiles = (n_nodes + 15) / 16;
    const int waves_per_block = 256 / 32;
    const int blocks1 = (n_tiles + waves_per_block - 1) / waves_per_block;
    node_scores_wmma_kernel<<<blocks1, 256, 0, stream>>>(
        x, w_src, w_tgt, s_src, s_tgt, n_nodes, n_tiles);

    const int blocks2 = (n_edges + 255) / 256;
    edge_gate_kernel<<<blocks2, 256, 0, stream>>>(
        edge_index, s_src, s_tgt, out, n_edges);
}